// CausalLinearAttention_70111046140053
// MI455X (gfx1250) — compile-verified
//
#include <hip/hip_runtime.h>
#include <hip/hip_bf16.h>

typedef __attribute__((ext_vector_type(2))) float v2f;
typedef __attribute__((ext_vector_type(8))) float v8f;

#define SEQ_N   2
#define SEQ_L   2048
#define SEQ_H   8
#define DIM     64
#define NH      (SEQ_N * SEQ_H)          // 16 independent sequences
#define NSC     16                        // super-chunks per sequence
#define SC_LEN  (SEQ_L / NSC)             // 128 timesteps per super-chunk
#define CHUNKS  (SC_LEN / 16)             // 8 chunks of 16 per super-chunk
#define PART_STRIDE (DIM * DIM + DIM)     // 64x64 state + 64 ksum = 4160 floats
#define EPSV    1e-10f

__device__ __forceinline__ v8f wmma4(v2f a, v2f b, v8f c) {
    // V_WMMA_F32_16X16X4_F32 : D(16x16 f32) = A(16x4 f32) * B(4x16 f32) + C
    return __builtin_amdgcn_wmma_f32_16x16x4_f32(
        /*neg_a=*/false, a, /*neg_b=*/false, b,
        /*c_mod=*/(short)0, c, /*reuse_a=*/false, /*reuse_b=*/false);
}

__device__ __forceinline__ float fmap(float x) {   // elu(x) + 1
    return x > 0.0f ? x + 1.0f : __expf(x);
}

// Async global -> LDS copy of 16B per lane (CDNA5 GLOBAL_LOAD_ASYNC_TO_LDS_B128,
// GV addressing; VDST carries the wave-relative LDS byte offset = low 32 bits of
// the generic aperture address of a __shared__ pointer). Tracked by ASYNCcnt.
__device__ __forceinline__ void async_v_copy(const float* gsrc, float* ldst) {
    uint32_t loff = (uint32_t)(uintptr_t)ldst;
    asm volatile("global_load_async_to_lds_b128 %0, %1, off"
                 :: "v"(loff), "v"(gsrc) : "memory");
}
#define WAIT_ASYNC_LE2() asm volatile("s_wait_asynccnt 0x2" ::: "memory")
#define WAIT_ASYNC_0()   asm volatile("s_wait_asynccnt 0x0" ::: "memory")

// ---------------------------------------------------------------------------
// Phase 1: per (sequence, super-chunk) partial state: S = sum phi(k)^T v,
//          ksum = sum phi(k).  4 waves, each owns a 16-wide m-slice.
// ---------------------------------------------------------------------------
__global__ __launch_bounds__(128)
void la_phase1(const float* __restrict__ K, const float* __restrict__ V,
               float* __restrict__ part) {
    __shared__ float sK[16 * 68];
    __shared__ float sV[2][16 * 68];
    __shared__ float sKsum[64];

    const int sc   = blockIdx.x;
    const int nh   = blockIdx.y;
    const int tid  = threadIdx.x;
    const int wave = tid >> 5;
    const int lane = tid & 31;
    const int half = lane >> 4;
    const int lc   = lane & 15;
    const int m0   = wave * 16;

    const size_t base = (size_t)((nh >> 3) * SEQ_L * SEQ_H + (nh & 7)) * DIM;
    const int    l0   = sc * SC_LEN;

    const int r  = tid >> 3;        // 0..15 (chunk row this thread stages)
    const int c8 = (tid & 7) * 8;   // 0..56

    // prologue: async-stage chunk 0 of V into buffer 0
    {
        const float* vp = V + base + (size_t)(l0 + r) * (SEQ_H * DIM) + c8;
        async_v_copy(vp,     &sV[0][r * 68 + c8]);
        async_v_copy(vp + 4, &sV[0][r * 68 + c8 + 4]);
    }

    v8f S[4];
    v8f zero = {0.f, 0.f, 0.f, 0.f, 0.f, 0.f, 0.f, 0.f};
#pragma unroll
    for (int t = 0; t < 4; t++) S[t] = zero;
    if (tid < 64) sKsum[tid] = 0.0f;
    __syncthreads();

    for (int ch = 0; ch < CHUNKS; ch++) {
        const int vb = ch & 1;
        const bool pre = (ch + 1 < CHUNKS);
        // issue async copy of NEXT chunk's V into the other buffer
        if (pre) {
            const float* vp = V + base + (size_t)(l0 + (ch + 1) * 16 + r) * (SEQ_H * DIM) + c8;
            async_v_copy(vp,     &sV[vb ^ 1][r * 68 + c8]);
            async_v_copy(vp + 4, &sV[vb ^ 1][r * 68 + c8 + 4]);
        }
        // synchronous stage of K (feature map applied) + prefetch next K rows
        {
            const float* kp = K + base + (size_t)(l0 + ch * 16 + r) * (SEQ_H * DIM) + c8;
#pragma unroll
            for (int j = 0; j < 8; j++) sK[r * 68 + c8 + j] = fmap(kp[j]);
            if (pre) __builtin_prefetch(kp + 16 * SEQ_H * DIM, 0, 3);
        }
        if (pre) { WAIT_ASYNC_LE2(); } else { WAIT_ASYNC_0(); }
        __syncthreads();

        // S_t += Kf^T(16 d-rows x 16 steps) @ V(16 steps x 16 m)  [4 k-steps of K=4]
#pragma unroll
        for (int t = 0; t < 4; t++) {
            v8f s = S[t];
#pragma unroll
            for (int kt = 0; kt < 4; kt++) {
                const int j0 = 4 * kt + half * 2;
                v2f a, b;
                a.x = sK[j0 * 68 + 16 * t + lc];
                a.y = sK[(j0 + 1) * 68 + 16 * t + lc];
                b.x = sV[vb][j0 * 68 + m0 + lc];
                b.y = sV[vb][(j0 + 1) * 68 + m0 + lc];
                s = wmma4(a, b, s);
            }
            S[t] = s;
        }
        // ksum accumulate (one owner thread per d)
        if (tid < 64) {
            float s = sKsum[tid];
#pragma unroll
            for (int j = 0; j < 16; j++) s += sK[j * 68 + tid];
            sKsum[tid] = s;
        }
        __syncthreads();
    }

    // write partial: S[d][m] row-major + ksum
    float* pp = part + (size_t)(nh * NSC + sc) * PART_STRIDE;
#pragma unroll
    for (int t = 0; t < 4; t++) {
#pragma unroll
        for (int e = 0; e < 8; e++) {
            const int d = 16 * t + e + half * 8;
            pp[d * DIM + m0 + lc] = S[t][e];
        }
    }
    if (tid < 64) pp[DIM * DIM + tid] = sKsum[tid];
}

// ---------------------------------------------------------------------------
// Phase 2: exclusive prefix scan of the NSC partial states per sequence.
// ---------------------------------------------------------------------------
__global__ __launch_bounds__(256)
void la_phase2(const float* __restrict__ part, float* __restrict__ carry) {
    const int s = blockIdx.x;   // sequence (nh)
    for (int idx = threadIdx.x; idx < PART_STRIDE; idx += blockDim.x) {
        float run = 0.0f;
#pragma unroll 1
        for (int c = 0; c < NSC; c++) {
            const size_t o = (size_t)(s * NSC + c) * PART_STRIDE + idx;
            carry[o] = run;
            run += part[o];
        }
    }
}

// ---------------------------------------------------------------------------
// Phase 3: per (sequence, super-chunk): chunked causal linear attention with
//          carried-in state.  All matmuls via V_WMMA_F32_16X16X4_F32.
// ---------------------------------------------------------------------------
__global__ __launch_bounds__(128)
void la_phase3(const float* __restrict__ Q, const float* __restrict__ K,
               const float* __restrict__ V, const float* __restrict__ carry,
               float* __restrict__ out) {
    __shared__ float sQ[16 * 68];
    __shared__ float sK[16 * 68];
    __shared__ float sV[2][16 * 68];    // double-buffered, filled by async DMA
    __shared__ float sS[4 * 16 * 66];   // per-wave S mirror, [m-col][d], stride 66
    __shared__ float sSc[4 * 16 * 18];  // per-wave masked-score scratch, stride 18
    __shared__ float sKsum[64];

    const int sc   = blockIdx.x;
    const int nh   = blockIdx.y;
    const int tid  = threadIdx.x;
    const int wave = tid >> 5;
    const int lane = tid & 31;
    const int half = lane >> 4;
    const int lc   = lane & 15;
    const int m0   = wave * 16;

    const size_t base = (size_t)((nh >> 3) * SEQ_L * SEQ_H + (nh & 7)) * DIM;
    const int    l0   = sc * SC_LEN;

    const int r  = tid >> 3;
    const int c8 = (tid & 7) * 8;

    // prologue: async-stage chunk 0 of V into buffer 0
    {
        const float* vp = V + base + (size_t)(l0 + r) * (SEQ_H * DIM) + c8;
        async_v_copy(vp,     &sV[0][r * 68 + c8]);
        async_v_copy(vp + 4, &sV[0][r * 68 + c8 + 4]);
    }

    // init carried state (registers + transposed LDS mirror) and ksum
    const float* cr = carry + (size_t)(nh * NSC + sc) * PART_STRIDE;
    v8f S[4];
#pragma unroll
    for (int t = 0; t < 4; t++) {
        v8f s;
#pragma unroll
        for (int e = 0; e < 8; e++) {
            const int d = 16 * t + e + half * 8;
            const float v = cr[d * DIM + m0 + lc];
            s[e] = v;
            sS[wave * (16 * 66) + lc * 66 + d] = v;
        }
        S[t] = s;
    }
    if (tid < 64) sKsum[tid] = cr[DIM * DIM + tid];
    __syncthreads();

    const v8f zero = {0.f, 0.f, 0.f, 0.f, 0.f, 0.f, 0.f, 0.f};

    for (int ch = 0; ch < CHUNKS; ch++) {
        const int vb = ch & 1;
        const bool pre = (ch + 1 < CHUNKS);
        // async copy of NEXT chunk's V into the other buffer (overlaps compute)
        if (pre) {
            const float* vp = V + base + (size_t)(l0 + (ch + 1) * 16 + r) * (SEQ_H * DIM) + c8;
            async_v_copy(vp,     &sV[vb ^ 1][r * 68 + c8]);
            async_v_copy(vp + 4, &sV[vb ^ 1][r * 68 + c8 + 4]);
        }
        // synchronous stage of Q,K (feature map) + prefetch of next chunk rows
        {
            const size_t row = base + (size_t)(l0 + ch * 16 + r) * (SEQ_H * DIM) + c8;
            const float* qp = Q + row;
            const float* kp = K + row;
#pragma unroll
            for (int j = 0; j < 8; j++) {
                sQ[r * 68 + c8 + j] = fmap(qp[j]);
                sK[r * 68 + c8 + j] = fmap(kp[j]);
            }
            if (pre) {
                __builtin_prefetch(qp + 16 * SEQ_H * DIM, 0, 3);
                __builtin_prefetch(kp + 16 * SEQ_H * DIM, 0, 3);
            }
        }
        if (pre) { WAIT_ASYNC_LE2(); } else { WAIT_ASYNC_0(); }
        __syncthreads();

        // scores = Qf @ Kf^T ; dvec = Qf @ broadcast(ksum)   (16 k-steps over d=64)
        v8f scr = zero, dv = zero;
#pragma unroll
        for (int kt = 0; kt < 16; kt++) {
            const int dd0 = 4 * kt + half * 2;
            v2f a  = *(const v2f*)&sQ[lc * 68 + dd0];
            v2f bk = *(const v2f*)&sK[lc * 68 + dd0];
            v2f bs = *(const v2f*)&sKsum[dd0];
            scr = wmma4(a, bk, scr);
            dv  = wmma4(a, bs, dv);
        }

        // causal mask (inclusive diagonal), row-sum -> denominator, stash scores
        float denom[8];
#pragma unroll
        for (int e = 0; e < 8; e++) {
            const int row = e + half * 8;
            const float v = (lc <= row) ? scr[e] : 0.0f;
            sSc[wave * (16 * 18) + row * 18 + lc] = v;
            float rs = v;
            rs += __shfl_xor(rs, 1, 32);
            rs += __shfl_xor(rs, 2, 32);
            rs += __shfl_xor(rs, 4, 32);
            rs += __shfl_xor(rs, 8, 32);
            denom[e] = dv[e] + rs + EPSV;
        }

        // out_inter = Qf @ S_prev (16 k-steps over d), read S from mirror
        v8f acc = zero;
#pragma unroll
        for (int kt = 0; kt < 16; kt++) {
            const int dd0 = 4 * kt + half * 2;
            v2f a = *(const v2f*)&sQ[lc * 68 + dd0];
            v2f b = *(const v2f*)&sS[wave * (16 * 66) + lc * 66 + dd0];
            acc = wmma4(a, b, acc);
        }
        // out_intra = masked_scores @ Vslice (4 k-steps over 16 timesteps)
#pragma unroll
        for (int kt = 0; kt < 4; kt++) {
            const int j0 = 4 * kt + half * 2;
            v2f a = *(const v2f*)&sSc[wave * (16 * 18) + lc * 18 + j0];
            v2f b;
            b.x = sV[vb][j0 * 68 + m0 + lc];
            b.y = sV[vb][(j0 + 1) * 68 + m0 + lc];
            acc = wmma4(a, b, acc);
        }

        // normalize + store
#pragma unroll
        for (int e = 0; e < 8; e++) {
            const int row = e + half * 8;
            out[base + (size_t)(l0 + ch * 16 + row) * (SEQ_H * DIM) + m0 + lc] =
                acc[e] / denom[e];
        }
        __syncthreads();   // everyone done reading sKsum before it is updated

        // state update: S_t += Kf^T @ V ; refresh LDS mirror
#pragma unroll
        for (int t = 0; t < 4; t++) {
            v8f s = S[t];
#pragma unroll
            for (int kt = 0; kt < 4; kt++) {
                const int j0 = 4 * kt + half * 2;
                v2f a, b;
                a.x = sK[j0 * 68 + 16 * t + lc];
                a.y = sK[(j0 + 1) * 68 + 16 * t + lc];
                b.x = sV[vb][j0 * 68 + m0 + lc];
                b.y = sV[vb][(j0 + 1) * 68 + m0 + lc];
                s = wmma4(a, b, s);
            }
            S[t] = s;
#pragma unroll
            for (int e = 0; e < 8; e++) {
                const int d = 16 * t + e + half * 8;
                sS[wave * (16 * 66) + lc * 66 + d] = s[e];
            }
        }
        if (tid < 64) {
            float s = sKsum[tid];
#pragma unroll
            for (int j = 0; j < 16; j++) s += sK[j * 68 + tid];
            sKsum[tid] = s;
        }
        __syncthreads();   // protect sQ/sK/sV/sKsum for next chunk
    }
}

// ---------------------------------------------------------------------------
extern "C" void kernel_launch(void* const* d_in, const int* in_sizes, int n_in,
                              void* d_out, int out_size, void* d_ws, size_t ws_size,
                              hipStream_t stream) {
    const float* Q = (const float*)d_in[0];
    const float* K = (const float*)d_in[1];
    const float* V = (const float*)d_in[2];
    float* out = (float*)d_out;

    float* part  = (float*)d_ws;                         // NH*NSC*PART_STRIDE
    float* carry = part + (size_t)NH * NSC * PART_STRIDE;

    dim3 grid(NSC, NH);
    dim3 blk(128);
    la_phase1<<<grid, blk, 0, stream>>>(K, V, part);
    la_phase2<<<NH, 256, 0, stream>>>(part, carry);
    la_phase3<<<grid, blk, 0, stream>>>(Q, K, V, carry, out);
}